// GCN_10754598109885
// MI455X (gfx1250) — compile-verified
//
#include <hip/hip_runtime.h>
#include <hip/hip_bf16.h>

typedef float v2f __attribute__((ext_vector_type(2)));
typedef float v8f __attribute__((ext_vector_type(8)));

#define IN_CH 512
#define HID   16
#define OUTC  64

// ---------------- degree / normalization ----------------

__global__ void k_init_deg(float* deg, int n) {
    int i = blockIdx.x * blockDim.x + threadIdx.x;
    if (i < n) deg[i] = 1.0f;               // self loop
}

__global__ void k_count_deg(const int* __restrict__ dst, float* deg, long long E) {
    long long e = (long long)blockIdx.x * blockDim.x + threadIdx.x;
    if (e < E) atomicAdd(&deg[dst[e]], 1.0f);
}

__global__ void k_rsqrt(float* deg, int n) {
    int i = blockIdx.x * blockDim.x + threadIdx.x;
    if (i < n) deg[i] = __frsqrt_rn(deg[i]);   // deg >= 1 always
}

// ---------------- GEMM1: H = X * W1   [n,512]x[512,16] ----------------
// One wave per 16-row tile, f32 WMMA 16x16x4, W1 staged in LDS.

__global__ __launch_bounds__(256) void k_gemm1(const float* __restrict__ x,
                                               const float* __restrict__ W1,
                                               float* __restrict__ h,
                                               int nTiles, int n) {
    __shared__ float sW[IN_CH * HID];                 // 32 KB
    for (int i = threadIdx.x; i < IN_CH * HID; i += 256) sW[i] = W1[i];
    __syncthreads();

    const int wave = threadIdx.x >> 5;
    const int lane = threadIdx.x & 31;
    const int tile = blockIdx.x * 8 + wave;
    if (tile >= nTiles) return;                       // wave-uniform: EXEC stays full

    const int m  = lane & 15;                         // A: row in tile / B: column
    const int kh = (lane >> 4) << 1;                  // lanes 0-15 -> K{0,1}, 16-31 -> K{2,3}
    int row = tile * 16 + m; if (row >= n) row = n - 1;
    const float* __restrict__ xrow = x + (size_t)row * IN_CH;

    v8f c = {0.f,0.f,0.f,0.f,0.f,0.f,0.f,0.f};
    for (int k = 0; k < IN_CH; k += 4) {
        v2f a = *(const v2f*)(xrow + k + kh);         // 8B aligned (k%4==0, kh even)
        v2f b;
        b.x = sW[(k + kh)     * HID + m];
        b.y = sW[(k + kh + 1) * HID + m];
        c = __builtin_amdgcn_wmma_f32_16x16x4_f32(false, a, false, b,
                                                  (short)0, c, false, false);
    }
    // C/D layout: VGPR r -> M = r (+8 for upper half-wave), N = lane&15
    const int mBase = tile * 16 + ((lane >> 4) << 3);
#pragma unroll
    for (int r = 0; r < 8; ++r) {
        int mr = mBase + r;
        if (mr < n) h[(size_t)mr * HID + m] = c[r];
    }
}

// ---------------- aggregation layer 1 (16 features) ----------------

__global__ void k_init_acc1(const float* __restrict__ h, const float* __restrict__ dinv,
                            float* __restrict__ acc, int n) {
    int t = blockIdx.x * blockDim.x + threadIdx.x;
    if (t >= n * HID) return;
    int i = t >> 4;
    float di = dinv[i];
    acc[t] = h[t] * di * di;                          // self-loop contribution
}

__global__ void k_edge1(const int* __restrict__ src, const int* __restrict__ dst,
                        const float* __restrict__ dinv, const float* __restrict__ h,
                        float* __restrict__ acc, long long nWork) {
    long long t = (long long)blockIdx.x * blockDim.x + threadIdx.x;
    if (t >= nWork) return;
    int f = (int)(t & (HID - 1));
    long long e = t >> 4;
    int s = src[e], d = dst[e];
    float w = dinv[s] * dinv[d];
    atomicAdd(&acc[(size_t)d * HID + f], h[(size_t)s * HID + f] * w);
}

__global__ void k_fin1(float* __restrict__ acc, const float* __restrict__ b1, int n) {
    int t = blockIdx.x * blockDim.x + threadIdx.x;
    if (t >= n * HID) return;
    float v = acc[t] + b1[t & (HID - 1)];
    acc[t] = v > 0.f ? v : 0.f;                       // in-place: acc1 becomes H1
}

// ---------------- GEMM2: H2 = H1 * W2   [n,16]x[16,64] ----------------
// One wave per (16-row, 16-col) tile; 4 col tiles; K=16 -> 4 WMMA steps.

__global__ __launch_bounds__(256) void k_gemm2(const float* __restrict__ h1,
                                               const float* __restrict__ W2,
                                               float* __restrict__ h2,
                                               int nWaveTiles, int n) {
    __shared__ float sW[HID * OUTC];                  // 4 KB
    for (int i = threadIdx.x; i < HID * OUTC; i += 256) sW[i] = W2[i];
    __syncthreads();

    const int wave = threadIdx.x >> 5;
    const int lane = threadIdx.x & 31;
    const int t = blockIdx.x * 8 + wave;
    if (t >= nWaveTiles) return;                      // wave-uniform
    const int rowTile = t >> 2;
    const int nTile   = t & 3;

    const int m  = lane & 15;
    const int kh = (lane >> 4) << 1;
    int row = rowTile * 16 + m; if (row >= n) row = n - 1;
    const float* __restrict__ arow = h1 + (size_t)row * HID;

    v8f c = {0.f,0.f,0.f,0.f,0.f,0.f,0.f,0.f};
#pragma unroll
    for (int k = 0; k < HID; k += 4) {
        v2f a = *(const v2f*)(arow + k + kh);
        v2f b;
        b.x = sW[(k + kh)     * OUTC + nTile * 16 + m];
        b.y = sW[(k + kh + 1) * OUTC + nTile * 16 + m];
        c = __builtin_amdgcn_wmma_f32_16x16x4_f32(false, a, false, b,
                                                  (short)0, c, false, false);
    }
    const int mBase = rowTile * 16 + ((lane >> 4) << 3);
#pragma unroll
    for (int r = 0; r < 8; ++r) {
        int mr = mBase + r;
        if (mr < n) h2[(size_t)mr * OUTC + nTile * 16 + m] = c[r];
    }
}

// ---------------- aggregation layer 2 (64 features) ----------------

__global__ void k_init_acc2(const float* __restrict__ h2, const float* __restrict__ dinv,
                            float* __restrict__ acc, int n) {
    long long t = (long long)blockIdx.x * blockDim.x + threadIdx.x;
    if (t >= (long long)n * OUTC) return;
    int i = (int)(t >> 6);
    float di = dinv[i];
    acc[t] = h2[t] * di * di;
}

__global__ void k_edge2(const int* __restrict__ src, const int* __restrict__ dst,
                        const float* __restrict__ dinv, const float* __restrict__ h2,
                        float* __restrict__ acc, long long nWork) {
    long long t = (long long)blockIdx.x * blockDim.x + threadIdx.x;
    if (t >= nWork) return;
    int f = (int)(t & (OUTC - 1));
    long long e = t >> 6;
    int s = src[e], d = dst[e];
    float w = dinv[s] * dinv[d];
    atomicAdd(&acc[(size_t)d * OUTC + f], h2[(size_t)s * OUTC + f] * w);
}

// ---------------- bias + log_softmax, one wave32 per node ----------------

__global__ __launch_bounds__(256) void k_fin2_softmax(const float* __restrict__ acc,
                                                      const float* __restrict__ b2,
                                                      float* __restrict__ out, int n) {
    const int wave = threadIdx.x >> 5;
    const int lane = threadIdx.x & 31;
    const int node = blockIdx.x * 8 + wave;
    if (node >= n) return;
    float v0 = acc[(size_t)node * OUTC + lane]      + b2[lane];
    float v1 = acc[(size_t)node * OUTC + lane + 32] + b2[lane + 32];
    float mx = fmaxf(v0, v1);
#pragma unroll
    for (int off = 16; off > 0; off >>= 1)
        mx = fmaxf(mx, __shfl_xor(mx, off, 32));
    float s = __expf(v0 - mx) + __expf(v1 - mx);
#pragma unroll
    for (int off = 16; off > 0; off >>= 1)
        s += __shfl_xor(s, off, 32);
    float lse = mx + __logf(s);
    out[(size_t)node * OUTC + lane]      = v0 - lse;
    out[(size_t)node * OUTC + lane + 32] = v1 - lse;
}

// ---------------- launcher ----------------

extern "C" void kernel_launch(void* const* d_in, const int* in_sizes, int n_in,
                              void* d_out, int out_size, void* d_ws, size_t ws_size,
                              hipStream_t stream) {
    const float* x  = (const float*)d_in[0];
    const int*   ei = (const int*)  d_in[1];   // [2, E] int32
    const float* W1 = (const float*)d_in[2];
    const float* b1 = (const float*)d_in[3];
    const float* W2 = (const float*)d_in[4];
    const float* b2 = (const float*)d_in[5];
    float* out = (float*)d_out;

    const int       n = in_sizes[0] / IN_CH;        // 100000
    const long long E = (long long)in_sizes[1] / 2; // 3200000
    const int* src = ei;
    const int* dst = ei + E;

    // workspace layout (floats)
    float* ws = (float*)d_ws;
    size_t o = 0;
    float* dinv = ws + o; o += ((size_t)n + 127) & ~(size_t)127;
    float* h1p  = ws + o; o += (size_t)n * HID;     // X*W1
    float* acc1 = ws + o; o += (size_t)n * HID;     // agg1 -> ReLU'd H1 (in place)
    float* h2p  = ws + o; o += (size_t)n * OUTC;    // H1*W2
    float* acc2 = ws + o; o += (size_t)n * OUTC;    // agg2

    const int nTiles = (n + 15) / 16;

    k_init_deg <<<(n + 255) / 256, 256, 0, stream>>>(dinv, n);
    k_count_deg<<<(unsigned)((E + 255) / 256), 256, 0, stream>>>(dst, dinv, E);
    k_rsqrt    <<<(n + 255) / 256, 256, 0, stream>>>(dinv, n);

    k_gemm1<<<(nTiles + 7) / 8, 256, 0, stream>>>(x, W1, h1p, nTiles, n);

    k_init_acc1<<<((size_t)n * HID + 255) / 256, 256, 0, stream>>>(h1p, dinv, acc1, n);
    {
        long long work = E * HID;
        k_edge1<<<(unsigned)((work + 255) / 256), 256, 0, stream>>>(src, dst, dinv, h1p, acc1, work);
    }
    k_fin1<<<((size_t)n * HID + 255) / 256, 256, 0, stream>>>(acc1, b1, n);

    k_gemm2<<<(nTiles * 4 + 7) / 8, 256, 0, stream>>>(acc1, W2, h2p, nTiles * 4, n);

    k_init_acc2<<<(unsigned)(((long long)n * OUTC + 255) / 256), 256, 0, stream>>>(h2p, dinv, acc2, n);
    {
        long long work = E * OUTC;
        k_edge2<<<(unsigned)((work + 255) / 256), 256, 0, stream>>>(src, dst, dinv, h2p, acc2, work);
    }
    k_fin2_softmax<<<(n + 7) / 8, 256, 0, stream>>>(acc2, b2, out, n);
}